// KorniaTrainAugmentation_86303072846061
// MI455X (gfx1250) — compile-verified
//
#include <hip/hip_runtime.h>
#include <stdint.h>
#include <stddef.h>

// ---------------------------------------------------------------------------
// Kornia-style train augmentation for MI455X (gfx1250).
// Pipeline: affine warp + bilinear -> coverage mask -> max-square DP crop ->
// bilinear resize 224^2 -> color jitter (HSV) -> separable 3x9 Gaussian blur.
// Bandwidth-bound (~250 MB total, L2-resident); blur implemented as banded
// matmuls on V_WMMA_F32_16X16X4_F32.
// ---------------------------------------------------------------------------

#define BATCH 64
#define CH    3
#define HH    256
#define WW    256
#define IMGSZ 224
#define NPIX_IN  (HH * WW)          // 65536
#define NPIX_OUT (IMGSZ * IMGSZ)    // 50176

typedef float v2f __attribute__((ext_vector_type(2)));
typedef float v8f __attribute__((ext_vector_type(8)));

struct BParam {
  float minv[6];                // inverse affine (2x3, last row [0 0 1])
  float fb, fc, fs, fhue;       // color jitter factors
  float sigma;                  // blur sigma
  float mean;                   // per-batch gray mean (filled by k_mean)
  int   applyColor;
  int   applyBlur;
  int   x0, y0, s;              // max-square crop (filled by k_maxsq)
  int   _pad;
};

// ------------------------- deterministic RNG -------------------------------
__device__ __forceinline__ uint64_t splitmix64(uint64_t z) {
  z += 0x9E3779B97F4A7C15ull;
  z = (z ^ (z >> 30)) * 0xBF58476D1CE4E5B9ull;
  z = (z ^ (z >> 27)) * 0x94D049BB133111EBull;
  return z ^ (z >> 31);
}
__device__ __forceinline__ float rngu(uint32_t stream, uint32_t i) {
  uint64_t h = splitmix64(((uint64_t)stream << 32) ^ (uint64_t)i ^ 0x2A2A2A2Aull);
  return (float)(h >> 40) * (1.0f / 16777216.0f);   // [0,1)
}

__device__ __forceinline__ float clip01(float v) {
  return fminf(fmaxf(v, 0.0f), 1.0f);
}
__device__ __forceinline__ int refl(int i, int n) {   // jnp.pad 'reflect'
  if (i < 0) i = -i;
  if (i >= n) i = 2 * n - 2 - i;
  return i;
}
__device__ __forceinline__ float band9(const float k[9], int d) {
  float r = 0.0f;
#pragma unroll
  for (int j = 0; j < 9; j++) r = (d == j) ? k[j] : r;
  return r;
}
__device__ __forceinline__ float band3(const float k[3], int d) {
  float r = 0.0f;
#pragma unroll
  for (int j = 0; j < 3; j++) r = (d == j) ? k[j] : r;
  return r;
}

// ------------------------- K0: per-batch parameters ------------------------
__global__ void k_params(BParam* __restrict__ prm, float* __restrict__ maxv) {
  int b = threadIdx.x;
  if (b == 0) *maxv = 0.0f;
  if (b >= BATCH) return;

  const float PI = 3.14159265358979323846f;
  bool  fh  = rngu(0, b) < 0.5f;
  bool  fv  = rngu(1, b) < 0.5f;
  float ang = (-20.0f + 40.0f * rngu(2, b)) * (PI / 180.0f);
  float sc  = 0.85f + 0.30f * rngu(3, b);
  float tx  = (-0.1f + 0.2f * rngu(4, b)) * (float)WW;
  float ty  = (-0.1f + 0.2f * rngu(5, b)) * (float)HH;

  float cx = (WW - 1) * 0.5f, cy = (HH - 1) * 0.5f;
  float ca = cosf(ang) * sc, sa = sinf(ang) * sc;
  float m02 = tx + cx - ca * cx + sa * cy;
  float m12 = ty + cy - sa * cx - ca * cy;

  float fx = fh ? -1.0f : 1.0f, fy = fv ? -1.0f : 1.0f;
  float ox = fh ? (float)(WW - 1) : 0.0f, oy = fv ? (float)(HH - 1) : 0.0f;
  // M = A @ F
  float M00 = ca * fx,  M01 = -sa * fy, M02 = ca * ox - sa * oy + m02;
  float M10 = sa * fx,  M11 =  ca * fy, M12 = sa * ox + ca * oy + m12;
  // inverse of affine 2x3
  float det = M00 * M11 - M01 * M10;
  float id  = 1.0f / det;
  BParam p;
  p.minv[0] =  M11 * id;
  p.minv[1] = -M01 * id;
  p.minv[2] = (M01 * M12 - M02 * M11) * id;
  p.minv[3] = -M10 * id;
  p.minv[4] =  M00 * id;
  p.minv[5] = (M02 * M10 - M00 * M12) * id;

  p.fb   = 0.7f + 0.6f * rngu(10, b);
  p.fc   = 0.7f + 0.6f * rngu(11, b);
  p.fs   = 0.7f + 0.6f * rngu(12, b);
  p.fhue = -0.15f + 0.30f * rngu(13, b);
  p.applyColor = rngu(14, b) < 0.7f;
  p.sigma = 0.1f + 1.9f * rngu(20, b);
  p.applyBlur = rngu(21, b) < 0.5f;
  p.mean = 0.0f; p.x0 = 0; p.y0 = 0; p.s = 1; p._pad = 0;
  prm[b] = p;
}

// ------------------------- K1: global max (for /255 heuristic) -------------
__global__ void k_maxred(const float* __restrict__ x, float* __restrict__ maxv, int n) {
  __shared__ float sm[256];
  float m = 0.0f;   // inputs are non-negative
  for (int i = blockIdx.x * blockDim.x + threadIdx.x; i < n; i += gridDim.x * blockDim.x)
    m = fmaxf(m, x[i]);
  sm[threadIdx.x] = m;
  __syncthreads();
  for (int s = 128; s > 0; s >>= 1) {
    if (threadIdx.x < s) sm[threadIdx.x] = fmaxf(sm[threadIdx.x], sm[threadIdx.x + s]);
    __syncthreads();
  }
  if (threadIdx.x == 0)
    atomicMax((int*)maxv, __float_as_int(sm[0]));  // valid for non-negative floats
}

// ------------------------- K2: affine warp + coverage ----------------------
__global__ void k_warp(const float* __restrict__ x, float* __restrict__ xw,
                       unsigned char* __restrict__ msk,
                       const BParam* __restrict__ prm, const float* __restrict__ maxv) {
  int idx = blockIdx.x * blockDim.x + threadIdx.x;
  if (idx >= BATCH * NPIX_IN) return;
  int b = idx / NPIX_IN, rem = idx % NPIX_IN;
  int py = rem / WW, px = rem % WW;
  const BParam p = prm[b];
  float scale = (*maxv > 1.0f) ? (1.0f / 255.0f) : 1.0f;

  float sx = p.minv[0] * (float)px + p.minv[1] * (float)py + p.minv[2];
  float sy = p.minv[3] * (float)px + p.minv[4] * (float)py + p.minv[5];

  // coverage: bilinear of ones, border=False (unclamped floor, in-bounds weights)
  {
    float x0f = floorf(sx), y0f = floorf(sy);
    int ix0 = (int)x0f, iy0 = (int)y0f;
    float wx = sx - x0f, wy = sy - y0f;
    float i00 = (iy0 >= 0 && iy0 < HH && ix0 >= 0     && ix0 < WW)     ? 1.0f : 0.0f;
    float i01 = (iy0 >= 0 && iy0 < HH && ix0 + 1 >= 0 && ix0 + 1 < WW) ? 1.0f : 0.0f;
    float i10 = (iy0 + 1 >= 0 && iy0 + 1 < HH && ix0 >= 0 && ix0 < WW) ? 1.0f : 0.0f;
    float i11 = (iy0 + 1 >= 0 && iy0 + 1 < HH && ix0 + 1 >= 0 && ix0 + 1 < WW) ? 1.0f : 0.0f;
    float cov = (1 - wy) * (1 - wx) * i00 + (1 - wy) * wx * i01 +
                wy * (1 - wx) * i10 + wy * wx * i11;
    msk[b * NPIX_IN + rem] = (cov > 0.5f) ? 1 : 0;
  }

  // channels: border=True (clamped coords)
  float ccx = fminf(fmaxf(sx, 0.0f), (float)(WW - 1));
  float ccy = fminf(fmaxf(sy, 0.0f), (float)(HH - 1));
  float x0f = floorf(ccx), y0f = floorf(ccy);
  int ix0 = (int)x0f, iy0 = (int)y0f;
  float wx = ccx - x0f, wy = ccy - y0f;
  int ix1 = min(ix0 + 1, WW - 1), iy1 = min(iy0 + 1, HH - 1);
#pragma unroll
  for (int c = 0; c < CH; c++) {
    const float* pc = x + ((size_t)(b * CH + c)) * NPIX_IN;
    float v00 = pc[iy0 * WW + ix0], v01 = pc[iy0 * WW + ix1];
    float v10 = pc[iy1 * WW + ix0], v11 = pc[iy1 * WW + ix1];
    float v = v00 * (1 - wy) * (1 - wx) + v01 * (1 - wy) * wx +
              v10 * wy * (1 - wx) + v11 * wy * wx;
    xw[((size_t)(b * CH + c)) * NPIX_IN + rem] = v * scale;
  }
}

// ------------------------- K3: max-square DP (1 thread/batch, LDS rows) ----
__global__ void k_maxsq(const unsigned char* __restrict__ msk, BParam* __restrict__ prm) {
  __shared__ int prev[BATCH][WW];     // 64 KB of 320 KB LDS
  int b = threadIdx.x;
  if (b >= BATCH) return;
  for (int j = 0; j < WW; j++) prev[b][j] = 0;
  const unsigned char* m = msk + (size_t)b * NPIX_IN;
  int best = 0, bi = 0, bj = 0, cnt = 0;
  for (int i = 0; i < HH; i++) {
    int prevleft = 0, left = 0;
    for (int j = 0; j < WW; j++) {
      int pj = prev[b][j];
      int bb = min(pj, prevleft) + 1;
      int mv = (int)m[i * WW + j];
      int d = mv ? min(bb, left + 1) : 0;
      prev[b][j] = d;
      prevleft = pj;
      left = d;
      cnt += mv;
      if (d > best) { best = d; bi = i; bj = j; }   // first-occurrence argmax
    }
  }
  int s = best, x0 = bj + 1 - s, y0 = bi + 1 - s;
  if (best == 0) { x0 = WW / 2; y0 = HH / 2; s = 1; }
  if (cnt == NPIX_IN) { x0 = 0; y0 = 0; s = HH; }
  prm[b].x0 = x0; prm[b].y0 = y0; prm[b].s = s;
}

// ------------------------- K4: crop + resize (bilinear, border=True) -------
__global__ void k_crop(const float* __restrict__ xw, float* __restrict__ xc,
                       const BParam* __restrict__ prm) {
  int idx = blockIdx.x * blockDim.x + threadIdx.x;
  if (idx >= BATCH * NPIX_OUT) return;
  int b = idx / NPIX_OUT, rem = idx % NPIX_OUT;
  int v = rem / IMGSZ, u = rem % IMGSZ;
  const BParam p = prm[b];
  float sf = (float)p.s / (float)(IMGSZ - 1);
  float cxs = (float)p.x0 + sf * (float)u;
  float cys = (float)p.y0 + sf * (float)v;
  cxs = fminf(fmaxf(cxs, 0.0f), (float)(WW - 1));
  cys = fminf(fmaxf(cys, 0.0f), (float)(HH - 1));
  float x0f = floorf(cxs), y0f = floorf(cys);
  int ix0 = (int)x0f, iy0 = (int)y0f;
  float wx = cxs - x0f, wy = cys - y0f;
  int ix1 = min(ix0 + 1, WW - 1), iy1 = min(iy0 + 1, HH - 1);
  __builtin_prefetch(xw + ((size_t)(b * CH)) * NPIX_IN + iy0 * WW + ix0, 0, 0);
#pragma unroll
  for (int c = 0; c < CH; c++) {
    const float* pc = xw + ((size_t)(b * CH + c)) * NPIX_IN;
    float v00 = pc[iy0 * WW + ix0], v01 = pc[iy0 * WW + ix1];
    float v10 = pc[iy1 * WW + ix0], v11 = pc[iy1 * WW + ix1];
    float val = v00 * (1 - wy) * (1 - wx) + v01 * (1 - wy) * wx +
                v10 * wy * (1 - wx) + v11 * wy * wx;
    xc[((size_t)(b * CH + c)) * NPIX_OUT + rem] = val;
  }
}

// ------------------------- K5: per-batch gray mean (post-brightness) -------
__global__ void k_mean(const float* __restrict__ xc, BParam* __restrict__ prm) {
  __shared__ float sm[256];
  int b = blockIdx.x;
  float fb = prm[b].fb;
  const float* pr = xc + ((size_t)(b * CH + 0)) * NPIX_OUT;
  const float* pg = xc + ((size_t)(b * CH + 1)) * NPIX_OUT;
  const float* pb = xc + ((size_t)(b * CH + 2)) * NPIX_OUT;
  float acc = 0.0f;
  for (int i = threadIdx.x; i < NPIX_OUT; i += 256) {
    float r = clip01(pr[i] * fb), g = clip01(pg[i] * fb), bl = clip01(pb[i] * fb);
    acc += 0.299f * r + 0.587f * g + 0.114f * bl;
  }
  sm[threadIdx.x] = acc;
  __syncthreads();
  for (int s = 128; s > 0; s >>= 1) {
    if (threadIdx.x < s) sm[threadIdx.x] += sm[threadIdx.x + s];
    __syncthreads();
  }
  if (threadIdx.x == 0) prm[b].mean = sm[0] / (float)NPIX_OUT;
}

// ------------------------- K6: color jitter (in place on xc) ---------------
__global__ void k_jitter(float* __restrict__ xc, const BParam* __restrict__ prm) {
  int idx = blockIdx.x * blockDim.x + threadIdx.x;
  if (idx >= BATCH * NPIX_OUT) return;
  int b = idx / NPIX_OUT, rem = idx % NPIX_OUT;
  const BParam p = prm[b];
  if (!p.applyColor) return;   // keep original pixel

  float* pr = xc + ((size_t)(b * CH + 0)) * NPIX_OUT + rem;
  float* pg = xc + ((size_t)(b * CH + 1)) * NPIX_OUT + rem;
  float* pb = xc + ((size_t)(b * CH + 2)) * NPIX_OUT + rem;
  float r = *pr, g = *pg, bl = *pb;

  // brightness
  r = clip01(r * p.fb); g = clip01(g * p.fb); bl = clip01(bl * p.fb);
  // contrast (mean of brightness-adjusted gray)
  r  = clip01(p.fc * r  + (1.0f - p.fc) * p.mean);
  g  = clip01(p.fc * g  + (1.0f - p.fc) * p.mean);
  bl = clip01(p.fc * bl + (1.0f - p.fc) * p.mean);
  // saturation
  float gray = 0.299f * r + 0.587f * g + 0.114f * bl;
  r  = clip01(p.fs * r  + (1.0f - p.fs) * gray);
  g  = clip01(p.fs * g  + (1.0f - p.fs) * gray);
  bl = clip01(p.fs * bl + (1.0f - p.fs) * gray);
  // hue via HSV round trip
  const float eps = 1e-8f;
  float maxc = fmaxf(r, fmaxf(g, bl));
  float minc = fminf(r, fminf(g, bl));
  float d = maxc - minc;
  float sat = d / fmaxf(maxc, eps);
  float dd = fmaxf(d, eps);
  float h;
  if (maxc == r)      { h = fmodf((g - bl) / dd, 6.0f); if (h < 0.0f) h += 6.0f; }
  else if (maxc == g) { h = (bl - r) / dd + 2.0f; }
  else                { h = (r - g) / dd + 4.0f; }
  h *= (1.0f / 6.0f);
  if (d <= eps) h = 0.0f;
  h += p.fhue;

  float hf = h - floorf(h);
  float h6 = hf * 6.0f;
  int i = (int)floorf(h6);
  i = max(0, min(5, i));
  float f = h6 - (float)i;
  float vv = maxc;
  float pp = vv * (1.0f - sat);
  float qq = vv * (1.0f - sat * f);
  float tt = vv * (1.0f - sat * (1.0f - f));
  float ro, go, bo;
  switch (i) {
    case 0: ro = vv; go = tt; bo = pp; break;
    case 1: ro = qq; go = vv; bo = pp; break;
    case 2: ro = pp; go = vv; bo = tt; break;
    case 3: ro = pp; go = qq; bo = vv; break;
    case 4: ro = tt; go = pp; bo = vv; break;
    default: ro = vv; go = pp; bo = qq; break;
  }
  *pr = clip01(ro); *pg = clip01(go); *pb = clip01(bo);
}

// ------------------------- K7: separable Gaussian blur via WMMA ------------
// Per 16x16 output tile of one (b,c) image:
//   horizontal: OH(18x16) = LIN(18x24) . KX(24x16)   (two 16-row WMMA groups)
//   vertical:   OUT(16x16) = KY(16x18) . OH(18x16)   (K padded to 20)
// using V_WMMA_F32_16X16X4_F32 (wave32, one wave per block).
__global__ void k_blur(const float* __restrict__ xc, float* __restrict__ out,
                       const BParam* __restrict__ prm) {
  __shared__ float LIN[18][25];   // padded input tile (halo 1 vert, 4 horiz)
  __shared__ float OH[18][17];    // horizontal-pass result

  const int tile = blockIdx.x;                 // 0..195 (14x14 tiles)
  const int c = blockIdx.y;
  const int b = blockIdx.z;
  const int ty = (tile / 14) * 16;
  const int tx = (tile % 14) * 16;
  const int lane = threadIdx.x;                // 0..31

  const float sigma = prm[b].sigma;
  const int applyB = prm[b].applyBlur;
  const float ns2 = -1.0f / (2.0f * sigma * sigma);

  float ky[3], kx[9];
  {
    float sum = 0.0f;
#pragma unroll
    for (int i = 0; i < 3; i++) { float t = (float)(i - 1); ky[i] = expf(t * t * ns2); sum += ky[i]; }
    float inv = 1.0f / sum;
#pragma unroll
    for (int i = 0; i < 3; i++) ky[i] *= inv;
  }
  {
    float sum = 0.0f;
#pragma unroll
    for (int i = 0; i < 9; i++) { float t = (float)(i - 4); kx[i] = expf(t * t * ns2); sum += kx[i]; }
    float inv = 1.0f / sum;
#pragma unroll
    for (int i = 0; i < 9; i++) kx[i] *= inv;
  }

  const float* src = xc + ((size_t)(b * CH + c)) * NPIX_OUT;
  for (int t = lane; t < 18 * 24; t += 32) {
    int r = t / 24, k = t % 24;
    int gy = refl(ty + r - 1, IMGSZ);
    int gx = refl(tx + k - 4, IMGSZ);
    LIN[r][k] = src[gy * IMGSZ + gx];
  }
  __syncthreads();

  const int m  = lane & 15;          // row (A) / col (B,C,D)
  const int kb = (lane >> 4) << 1;   // K-half select (0 or 2)

  // Horizontal pass: rows 0..15 (oh0) and rows 2..17 (oh1).
  v8f oh0 = {};
  v8f oh1 = {};
#pragma unroll
  for (int chk = 0; chk < 6; chk++) {
    const int k0 = chk * 4;
    v2f bf;
    bf.x = band9(kx, (k0 + kb)     - m);   // KX[k][n] = kx[k-n]
    bf.y = band9(kx, (k0 + kb + 1) - m);
    v2f a0; a0.x = LIN[m][k0 + kb];     a0.y = LIN[m][k0 + kb + 1];
    oh0 = __builtin_amdgcn_wmma_f32_16x16x4_f32(false, a0, false, bf, (short)0, oh0, false, false);
    v2f a1; a1.x = LIN[m + 2][k0 + kb]; a1.y = LIN[m + 2][k0 + kb + 1];
    oh1 = __builtin_amdgcn_wmma_f32_16x16x4_f32(false, a1, false, bf, (short)0, oh1, false, false);
  }

  const int rbase = (lane >> 4) * 8;   // D layout: VGPR i <-> rows rbase+i
#pragma unroll
  for (int i = 0; i < 8; i++) OH[rbase + i][m] = oh0[i];
#pragma unroll
  for (int i = 0; i < 8; i++) {
    int r = 2 + rbase + i;
    if (r >= 16) OH[r][m] = oh1[i];    // only rows 16,17 are new
  }
  __syncthreads();

  // Vertical pass: OUT = KY(16x18) . OH(18x16), K padded to 20.
  v8f acc = {};
#pragma unroll
  for (int chk = 0; chk < 5; chk++) {
    const int k0 = chk * 4;
    v2f a;
    a.x = band3(ky, (k0 + kb)     - m);   // KY[r][k] = ky[k-r]
    a.y = band3(ky, (k0 + kb + 1) - m);
    int kr = k0 + kb;
    int kr0 = min(kr, 17), kr1 = min(kr + 1, 17);
    float t0 = OH[kr0][m], t1 = OH[kr1][m];   // unconditional loads, select after
    v2f bf;
    bf.x = (kr     < 18) ? t0 : 0.0f;
    bf.y = (kr + 1 < 18) ? t1 : 0.0f;
    acc = __builtin_amdgcn_wmma_f32_16x16x4_f32(false, a, false, bf, (short)0, acc, false, false);
  }

#pragma unroll
  for (int i = 0; i < 8; i++) {
    int r = rbase + i;
    float passv = LIN[r + 1][m + 4];          // un-blurred center pixel
    float v = applyB ? acc[i] : passv;
    out[((size_t)(b * CH + c) * IMGSZ + (ty + r)) * IMGSZ + (tx + m)] = clip01(v);
  }
}

// ------------------------- launcher ----------------------------------------
extern "C" void kernel_launch(void* const* d_in, const int* in_sizes, int n_in,
                              void* d_out, int out_size, void* d_ws, size_t ws_size,
                              hipStream_t stream) {
  const float* x = (const float*)d_in[0];
  float* out = (float*)d_out;

  // Workspace layout (bytes):
  //   [0)        BParam[64]
  //   [16384)    float maxval
  //   [32768)    xw   64*3*256*256 f32  (50331648 B)
  //   [50364416) mask 64*256*256 u8     ( 4194304 B)
  //   [54558720) xc   64*3*224*224 f32  (38535168 B)  -> total ~88.8 MB
  char* w = (char*)d_ws;
  BParam* prm        = (BParam*)(w + 0);
  float* maxv        = (float*)(w + 16384);
  float* xw          = (float*)(w + 32768);
  unsigned char* msk = (unsigned char*)(w + 50364416);
  float* xc          = (float*)(w + 54558720);

  k_params<<<1, 64, 0, stream>>>(prm, maxv);
  k_maxred<<<2048, 256, 0, stream>>>(x, maxv, BATCH * CH * NPIX_IN);
  k_warp<<<(BATCH * NPIX_IN + 255) / 256, 256, 0, stream>>>(x, xw, msk, prm, maxv);
  k_maxsq<<<1, 64, 0, stream>>>(msk, prm);
  k_crop<<<(BATCH * NPIX_OUT + 255) / 256, 256, 0, stream>>>(xw, xc, prm);
  k_mean<<<BATCH, 256, 0, stream>>>(xc, prm);
  k_jitter<<<(BATCH * NPIX_OUT + 255) / 256, 256, 0, stream>>>(xc, prm);
  k_blur<<<dim3(196, CH, BATCH), 32, 0, stream>>>(xc, out, prm);
}